// CausalGraphLayer_87926570484017
// MI455X (gfx1250) — compile-verified
//
#include <hip/hip_runtime.h>
#include <hip/hip_bf16.h>
#include <math.h>

// Problem constants (match reference)
#define BB 4
#define NN 1024
#define TT 64
#define CC 64
#define KK 16
#define MAXK 32
#define NB 4

typedef __attribute__((ext_vector_type(2))) float v2f;
typedef __attribute__((ext_vector_type(4))) float v4f;
typedef __attribute__((ext_vector_type(8))) float v8f;

// ---------------------------------------------------------------------------
// Phase 1: edge_weights[n][k][c] = (sum_beta coeff[c,beta]*bases[beta,n,k]) * adj[n,k]
// Real GEMM per node: (64x16) = (64x4) @ (4x16)  ->  V_WMMA_F32_16X16X4_F32
// (adj folds into the B operand since it does not depend on beta.)
// Grid: 1024 blocks (one node each), 128 threads = 4 waves = 4 c-tiles of 16.
// ---------------------------------------------------------------------------
__global__ __launch_bounds__(128)
void cgl_edge_weights_wmma(const float* __restrict__ adj,     // (N, MAXK)
                           const float* __restrict__ bw,      // (NB, N, MAXK)
                           const float* __restrict__ coeff,   // (C, NB)
                           float* __restrict__ ew)            // (N, KK, CC) scratch
{
    const int n    = blockIdx.x;
    const int wave = threadIdx.x >> 5;   // c-tile 0..3
    const int lane = threadIdx.x & 31;
    const int g    = lane >> 4;          // half-wave group
    const int l    = lane & 15;
    const int c0   = wave * 16;

#if __has_builtin(__builtin_amdgcn_wmma_f32_16x16x4_f32)
    // A tile (16x4): A[m=l, beta = j + 2g] = coeff[c0+m, beta]
    v2f a, b;
    a.x = coeff[(c0 + l) * NB + (2 * g + 0)];
    a.y = coeff[(c0 + l) * NB + (2 * g + 1)];
    // B tile (4x16): B[beta = j + 2g, k = l] = bases[beta, n, k] * adj[n, k]
    const float adj_k = adj[n * MAXK + l];
    b.x = bw[(size_t)(2 * g + 0) * (NN * MAXK) + n * MAXK + l] * adj_k;
    b.y = bw[(size_t)(2 * g + 1) * (NN * MAXK) + n * MAXK + l] * adj_k;

    v8f d = {};
    // 8 args: (neg_a, A, neg_b, B, c_mod, C, reuse_a, reuse_b)
    d = __builtin_amdgcn_wmma_f32_16x16x4_f32(false, a, false, b,
                                              (short)0, d, false, false);

    // D (16x16): lane L, vgpr v -> row M = v + 8g (channel), col N = l (k)
    // Per-lane D elements are contiguous in ew -> two global_store_b128.
    float* dst = ew + (size_t)n * (KK * CC);
#pragma unroll
    for (int v = 0; v < 8; ++v) {
        const int m = v + 8 * g;                 // channel within tile
        dst[l * CC + (c0 + m)] = d[v];           // layout [n][k][c]
    }
#else
    // Scalar fallback (should not be taken on gfx1250)
    for (int k = l; k < KK; k += 16) {
        for (int m = g; m < 16; m += 2) {
            float acc = 0.f;
            for (int be = 0; be < NB; ++be)
                acc += coeff[(c0 + m) * NB + be] *
                       bw[(size_t)be * (NN * MAXK) + n * MAXK + k];
            ew[(size_t)n * (KK * CC) + k * CC + (c0 + m)] = acc * adj[n * MAXK + k];
        }
    }
#endif
}

// ---------------------------------------------------------------------------
// Phase 2: out[b,n,t,c] = tanh( sum_k z[b, idx[n,k], t, c] * ew[n,k,c] )
// One block per (b,n); 256 threads. Each lane owns FOUR consecutive channels
// (cq = (tid&15)*4) and 4 t-rows -> every z access is a 16B global_load_b128
// (512 B per wave-instruction, 4x fewer VMEM ops than the b32 version).
// z (64 MB) is the 16x-reused working set and stays resident in the 192 MB
// L2; the write-once output is stored with non-temporal hints so it does not
// evict z. Addressing uses a uniform SGPR batch base + per-lane 32-bit
// element offsets (GVS mode) to minimize address VGPRs.
// ---------------------------------------------------------------------------
__global__ __launch_bounds__(256)
void cgl_aggregate(const float* __restrict__ z,      // (B, N, T, C)
                   const int*   __restrict__ nidx,   // (N, KK)
                   const float* __restrict__ ew,     // (N, KK, CC)
                   float* __restrict__ out)          // (B, N, T, C)
{
    __shared__ float s_w[KK * CC];   // 4 KB, layout [k][c]
    __shared__ int   s_off[KK];      // neighbor element offsets within batch

    const int bn  = blockIdx.x;          // b*N + n
    const int n   = bn & (NN - 1);
    const int b   = bn >> 10;            // NN == 1024
    const int tid = threadIdx.x;

    if (tid < KK) s_off[tid] = nidx[n * KK + tid] * (TT * CC);
#pragma unroll
    for (int i = 0; i < (KK * CC) / 256; ++i)
        s_w[tid + i * 256] = ew[(size_t)n * (KK * CC) + tid + i * 256];
    __syncthreads();

    const int cq = (tid & 15) * 4;       // quad of channels
    const int t0 = tid >> 4;             // 0..15

    const float* zb = z + (size_t)b * ((size_t)NN * TT * CC);  // uniform -> SGPR base

    v4f w[KK];
    int off[KK];
#pragma unroll
    for (int k = 0; k < KK; ++k) {
        w[k]   = *(const v4f*)&s_w[k * CC + cq];
        off[k] = s_off[k] + cq;          // 32-bit element offset (max ~4.2M, fits)
    }

    float* op = out + (size_t)bn * (TT * CC) + cq;

#pragma unroll 1
    for (int t = t0; t < TT; t += 16) {
        v4f acc = {0.f, 0.f, 0.f, 0.f};
#pragma unroll
        for (int k = 0; k < KK; ++k) {
            const v4f zv = *(const v4f*)(zb + off[k] + t * CC);  // global_load_b128
            acc.x = fmaf(zv.x, w[k].x, acc.x);
            acc.y = fmaf(zv.y, w[k].y, acc.y);
            acc.z = fmaf(zv.z, w[k].z, acc.z);
            acc.w = fmaf(zv.w, w[k].w, acc.w);
        }
        v4f r;
        r.x = tanhf(acc.x);
        r.y = tanhf(acc.y);
        r.z = tanhf(acc.z);
        r.w = tanhf(acc.w);
        __builtin_nontemporal_store(r, (v4f*)(op + t * CC));     // NT b128 store
    }
}

extern "C" void kernel_launch(void* const* d_in, const int* in_sizes, int n_in,
                              void* d_out, int out_size, void* d_ws, size_t ws_size,
                              hipStream_t stream)
{
    const float* z     = (const float*)d_in[0];   // (B,N,T,C) f32
    const int*   nidx  = (const int*)  d_in[1];   // (N,16)   i32
    const float* adj   = (const float*)d_in[2];   // (N,32)   f32
    const float* bw    = (const float*)d_in[3];   // (4,N,32) f32
    const float* coeff = (const float*)d_in[4];   // (64,4)   f32
    float*       out   = (float*)d_out;
    float*       ew    = (float*)d_ws;            // (N,16,64) f32 = 4 MB

    cgl_edge_weights_wmma<<<NN, 128, 0, stream>>>(adj, bw, coeff, ew);
    cgl_aggregate<<<BB * NN, 256, 0, stream>>>(z, nidx, ew, out);
}